// GNN_81724637708341
// MI455X (gfx1250) — compile-verified
//
#include <hip/hip_runtime.h>
#include <math.h>

typedef __attribute__((ext_vector_type(2))) float v2f;
typedef __attribute__((ext_vector_type(8))) float v8f;

#define DIM_IN  128
#define DIM_HID 64
#define DIM_OUT 40
#define DIM_OUT_PAD 48

// ---------------------------------------------------------------------------
// GEMM1: h1[N,64] = x[N,128] @ W1[128,64], exact f32 via V_WMMA_F32_16X16X4_F32.
// One wave -> 16 rows x 64 cols. 8 waves/block -> 128 rows/block.
// LDS holds W1 K-pair interleaved: w1s[(k/2)*128 + n*2 + (k&1)] = W1[k][n],
// so each WMMA B operand (B[k][n], B[k+1][n]) is one aligned ds_load_b64.
// ---------------------------------------------------------------------------
__global__ void __launch_bounds__(256)
gemm1_kernel(const float* __restrict__ x, const float* __restrict__ W1,
             float* __restrict__ h1, int n_nodes) {
    __shared__ float w1s[DIM_IN * DIM_HID];  // 32 KB
    {
        const float4* W1v = (const float4*)W1;
        for (int i = threadIdx.x; i < DIM_IN * DIM_HID / 4; i += blockDim.x) {
            float4 v = W1v[i];
            int k = i >> 4;              // 16 float4 per 64-wide row
            int n = (i & 15) * 4;
            float* p = &w1s[(k >> 1) * (2 * DIM_HID) + n * 2 + (k & 1)];
            p[0] = v.x; p[2] = v.y; p[4] = v.z; p[6] = v.w;
        }
    }
    __syncthreads();

    const int wave = threadIdx.x >> 5;
    const int lane = threadIdx.x & 31;
    const int m0 = (blockIdx.x * 8 + wave) * 16;
    if (m0 >= n_nodes) return;               // wave-uniform: EXEC stays all-1s below

    const int half = lane >> 4;              // 0: K=k,k+1   1: K=k+2,k+3
    const int l16  = lane & 15;
    const float* arow = x + (size_t)(m0 + l16) * DIM_IN + half * 2;

    v8f acc0 = {}, acc1 = {}, acc2 = {}, acc3 = {};
    #pragma unroll 4
    for (int k = 0; k < DIM_IN; k += 4) {
        float2 av = *(const float2*)(arow + k);
        v2f a; a.x = av.x; a.y = av.y;
        const float* bbase = &w1s[((k >> 1) + half) * (2 * DIM_HID) + l16 * 2];
        v2f b0 = *(const v2f*)(bbase + 0);    // nt=0: cols 0..15
        v2f b1 = *(const v2f*)(bbase + 32);   // nt=1: cols 16..31
        v2f b2 = *(const v2f*)(bbase + 64);   // nt=2: cols 32..47
        v2f b3 = *(const v2f*)(bbase + 96);   // nt=3: cols 48..63
        acc0 = __builtin_amdgcn_wmma_f32_16x16x4_f32(false, a, false, b0, (short)0, acc0, false, false);
        acc1 = __builtin_amdgcn_wmma_f32_16x16x4_f32(false, a, false, b1, (short)0, acc1, false, false);
        acc2 = __builtin_amdgcn_wmma_f32_16x16x4_f32(false, a, false, b2, (short)0, acc2, false, false);
        acc3 = __builtin_amdgcn_wmma_f32_16x16x4_f32(false, a, false, b3, (short)0, acc3, false, false);
    }

    // store: vgpr r, lane<16 -> (M=m0+r, N=l16+nt*16); lane>=16 -> M=m0+r+8
    const int mbase = m0 + half * 8;
    #pragma unroll
    for (int r = 0; r < 8; ++r) {
        float* orow = h1 + (size_t)(mbase + r) * DIM_HID + l16;
        orow[0]  = acc0[r];
        orow[16] = acc1[r];
        orow[32] = acc2[r];
        orow[48] = acc3[r];
    }
}

// ---------------------------------------------------------------------------
// SpMM over D=64: out[dst] += w * h[src]. 16 threads/edge, float4 per thread,
// hardware f32 atomics (global_atomic_add_f32). Accumulators live in L2.
// ---------------------------------------------------------------------------
__global__ void __launch_bounds__(256)
spmm64_kernel(const long long* __restrict__ ei, const float* __restrict__ ew,
              const float* __restrict__ h, float* __restrict__ out, int E) {
    long long tid = (long long)blockIdx.x * blockDim.x + threadIdx.x;
    if (tid >= (long long)E * 16) return;
    const int e = (int)(tid >> 4);
    const int c = ((int)tid & 15) * 4;
    const long long dst = ei[e];
    const long long src = ei[(size_t)E + e];
    const float w = ew[e];
    const float4 v = *(const float4*)(h + (size_t)src * DIM_HID + c);
    float* o = out + (size_t)dst * DIM_HID + c;
    unsafeAtomicAdd(o + 0, w * v.x);
    unsafeAtomicAdd(o + 1, w * v.y);
    unsafeAtomicAdd(o + 2, w * v.z);
    unsafeAtomicAdd(o + 3, w * v.w);
}

// ---------------------------------------------------------------------------
// GEMM2: h2[N,40] = relu(a1)[N,64] @ W2[64,40]; N padded 40->48 (3 WMMA tiles).
// Same K-pair interleaved LDS layout, 96 floats per kp row.
// ---------------------------------------------------------------------------
__global__ void __launch_bounds__(256)
gemm2_kernel(const float* __restrict__ a1, const float* __restrict__ W2,
             float* __restrict__ h2, int n_nodes) {
    __shared__ float w2s[(DIM_HID / 2) * (2 * DIM_OUT_PAD)];  // 12 KB
    for (int i = threadIdx.x; i < DIM_HID * DIM_OUT_PAD; i += blockDim.x) {
        int k = i / DIM_OUT_PAD, n = i % DIM_OUT_PAD;
        float v = (n < DIM_OUT) ? W2[k * DIM_OUT + n] : 0.0f;
        w2s[(k >> 1) * (2 * DIM_OUT_PAD) + n * 2 + (k & 1)] = v;
    }
    __syncthreads();

    const int wave = threadIdx.x >> 5;
    const int lane = threadIdx.x & 31;
    const int m0 = (blockIdx.x * 8 + wave) * 16;
    if (m0 >= n_nodes) return;

    const int half = lane >> 4;
    const int l16  = lane & 15;
    const float* arow = a1 + (size_t)(m0 + l16) * DIM_HID + half * 2;

    v8f acc0 = {}, acc1 = {}, acc2 = {};
    #pragma unroll 4
    for (int k = 0; k < DIM_HID; k += 4) {
        float2 av = *(const float2*)(arow + k);
        v2f a;                              // fused ReLU on the SpMM1 output
        a.x = fmaxf(av.x, 0.0f);
        a.y = fmaxf(av.y, 0.0f);
        const float* bbase = &w2s[((k >> 1) + half) * (2 * DIM_OUT_PAD) + l16 * 2];
        v2f b0 = *(const v2f*)(bbase + 0);
        v2f b1 = *(const v2f*)(bbase + 32);
        v2f b2 = *(const v2f*)(bbase + 64);
        acc0 = __builtin_amdgcn_wmma_f32_16x16x4_f32(false, a, false, b0, (short)0, acc0, false, false);
        acc1 = __builtin_amdgcn_wmma_f32_16x16x4_f32(false, a, false, b1, (short)0, acc1, false, false);
        acc2 = __builtin_amdgcn_wmma_f32_16x16x4_f32(false, a, false, b2, (short)0, acc2, false, false);
    }

    const int mbase = m0 + half * 8;
    #pragma unroll
    for (int r = 0; r < 8; ++r) {
        float* orow = h2 + (size_t)(mbase + r) * DIM_OUT + l16;
        orow[0]  = acc0[r];
        orow[16] = acc1[r];
        if (l16 < DIM_OUT - 32) orow[32] = acc2[r];   // cols 32..39 only
    }
}

// ---------------------------------------------------------------------------
// SpMM over D=40: 10 threads/edge, float4 per thread.
// ---------------------------------------------------------------------------
__global__ void __launch_bounds__(256)
spmm40_kernel(const long long* __restrict__ ei, const float* __restrict__ ew,
              const float* __restrict__ h, float* __restrict__ out, int E) {
    long long tid = (long long)blockIdx.x * blockDim.x + threadIdx.x;
    if (tid >= (long long)E * 10) return;
    const int e = (int)(tid / 10);
    const int c = (int)(tid - (long long)e * 10) * 4;
    const long long dst = ei[e];
    const long long src = ei[(size_t)E + e];
    const float w = ew[e];
    const float4 v = *(const float4*)(h + (size_t)src * DIM_OUT + c);
    float* o = out + (size_t)dst * DIM_OUT + c;
    unsafeAtomicAdd(o + 0, w * v.x);
    unsafeAtomicAdd(o + 1, w * v.y);
    unsafeAtomicAdd(o + 2, w * v.z);
    unsafeAtomicAdd(o + 3, w * v.w);
}

// ---------------------------------------------------------------------------
// Row-wise log_softmax over 40 classes; one wave32 per row (lane i covers
// element i and, for i<8, element 32+i). Wave shuffles for max / sum-exp.
// ---------------------------------------------------------------------------
__global__ void __launch_bounds__(256)
logsoftmax_kernel(const float* __restrict__ s, float* __restrict__ out, int n_nodes) {
    const int wave = threadIdx.x >> 5;
    const int lane = threadIdx.x & 31;
    const int row = blockIdx.x * 8 + wave;
    if (row >= n_nodes) return;
    const float* r = s + (size_t)row * DIM_OUT;
    float v0 = r[lane];
    float v1 = (lane < 8) ? r[32 + lane] : -INFINITY;
    float m = fmaxf(v0, v1);
    #pragma unroll
    for (int o = 16; o > 0; o >>= 1) m = fmaxf(m, __shfl_xor(m, o, 32));
    float se = expf(v0 - m) + ((lane < 8) ? expf(v1 - m) : 0.0f);
    #pragma unroll
    for (int o = 16; o > 0; o >>= 1) se += __shfl_xor(se, o, 32);
    const float lse = m + logf(se);
    float* orow = out + (size_t)row * DIM_OUT;
    orow[lane] = v0 - lse;
    if (lane < 8) orow[32 + lane] = v1 - lse;
}

// ---------------------------------------------------------------------------
extern "C" void kernel_launch(void* const* d_in, const int* in_sizes, int n_in,
                              void* d_out, int out_size, void* d_ws, size_t ws_size,
                              hipStream_t stream) {
    const float*     x  = (const float*)d_in[0];
    const long long* ei = (const long long*)d_in[1];
    const float*     ew = (const float*)d_in[2];
    const float*     W1 = (const float*)d_in[3];
    const float*     W2 = (const float*)d_in[4];
    float* out = (float*)d_out;

    const int N = in_sizes[0] / DIM_IN;   // 100000
    const int E = in_sizes[2];            // 3200000

    // workspace layout (floats): h1[N*64] | a1[N*64] | h2[N*40] | s2[N*40]
    float* h1 = (float*)d_ws;
    float* a1 = h1 + (size_t)N * DIM_HID;
    float* h2 = a1 + (size_t)N * DIM_HID;
    float* s2 = h2 + (size_t)N * DIM_OUT;

    hipMemsetAsync(a1, 0, (size_t)N * DIM_HID * sizeof(float), stream);
    hipMemsetAsync(s2, 0, (size_t)N * DIM_OUT * sizeof(float), stream);

    const int mBlocks = (N + 127) / 128;  // 128 rows per block (8 waves x 16)
    gemm1_kernel<<<mBlocks, 256, 0, stream>>>(x, W1, h1, N);

    long long t1 = (long long)E * 16;
    spmm64_kernel<<<(int)((t1 + 255) / 256), 256, 0, stream>>>(ei, ew, h1, a1, E);

    gemm2_kernel<<<mBlocks, 256, 0, stream>>>(a1, W2, h2, N);

    long long t2 = (long long)E * 10;
    spmm40_kernel<<<(int)((t2 + 255) / 256), 256, 0, stream>>>(ei, ew, h2, s2, E);

    logsoftmax_kernel<<<(N + 7) / 8, 256, 0, stream>>>(s2, out, N);
}